// ContrastiveLoss_10110353015496
// MI455X (gfx1250) — compile-verified
//
#include <hip/hip_runtime.h>

// ---------------------------------------------------------------------------
// ContrastiveLoss on MI455X (gfx1250)
//   Gram matrix via V_WMMA_F32_16X16X4_F32, operands staged through LDS with
//   double-buffered GLOBAL_LOAD_ASYNC_TO_LDS_B128 (ASYNCcnt-tracked DMA).
// B=4096, D=256, L=3, RADIUS=1.0
// ---------------------------------------------------------------------------

typedef float v2f __attribute__((ext_vector_type(2)));
typedef float v8f __attribute__((ext_vector_type(8)));

#define BN    4096
#define DK    256
#define TILE  128          // 128x128 output tile per workgroup (8 waves)
#define KB    16           // K-panel depth staged per async batch
#define LSTR  20           // LDS row stride in dwords (16 + 4 pad, conflict-free)

// Stage one 128 x KB f32 panel (row-major, global row stride DK) into LDS.
// 256 threads, each issues 2 async b128 transfers (4 float4 per row, 4
// threads per row -> coalesced 64B chunks). All LDS addresses 16B aligned.
__device__ __forceinline__ void stage_panel_async(const float* __restrict__ gbase,
                                                  unsigned ldsBase, int tid) {
#pragma unroll
    for (int it = 0; it < 2; ++it) {
        int e   = it * 256 + tid;        // 0..511 float4 chunks
        int row = e >> 2;                // 4 float4 per row
        int c4  = e & 3;
        const float* g = gbase + (size_t)row * DK + (c4 << 2);
        unsigned     l = ldsBase + (unsigned)(row * (LSTR * 4) + (c4 << 4));
        asm volatile("global_load_async_to_lds_b128 %0, %1, off"
                     :: "v"(l), "v"(g) : "memory");
    }
}

// ---------------------------------------------------------------------------
// Kernel 1: per-row squared norms + packed label keys (labels 0..7 -> exact
// 24-bit equality key). One wave per row.
// ---------------------------------------------------------------------------
__global__ __launch_bounds__(256) void prep_kernel(const float* __restrict__ F,
                                                   const int* __restrict__ labels,
                                                   float* __restrict__ sq,
                                                   unsigned* __restrict__ keys) {
    const int lane = threadIdx.x & 31;
    const int row  = blockIdx.x * 8 + (threadIdx.x >> 5);
    const float4* p = (const float4*)(F + (size_t)row * DK + lane * 8);
    float4 a = p[0];
    float4 b = p[1];
    float s = a.x * a.x + a.y * a.y + a.z * a.z + a.w * a.w +
              b.x * b.x + b.y * b.y + b.z * b.z + b.w * b.w;
    for (int off = 16; off >= 1; off >>= 1) s += __shfl_xor(s, off, 32);
    if (lane == 0) {
        sq[row] = s;
        int l0 = labels[row * 3 + 0];
        int l1 = labels[row * 3 + 1];
        int l2 = labels[row * 3 + 2];
        keys[row] = (unsigned)((l0 & 255) | ((l1 & 255) << 8) | ((l2 & 255) << 16));
    }
}

// ---------------------------------------------------------------------------
// Kernel 2: 128x128 Gram tiles. Double-buffered async LDS staging + f32 WMMA
// + fused contrastive epilogue. Grid 32x32, lower-triangular tiles exit.
//
// Fragment addressing for V_WMMA_F32_16X16X4_F32 (wave32):
//   A 16x4 : lane -> row = base + (lane&15), K = k0 + (lane>>4)*2 .. +1
//   B 4x16 : lane -> col = base + (lane&15), same K split
// Both are float2 reads from the row-major LDS panels (B panel = rows of F
// for the column block, since G = F F^T).
// ---------------------------------------------------------------------------
__global__ __launch_bounds__(256) void tile_kernel(const float* __restrict__ F,
                                                   const float* __restrict__ sq,
                                                   const unsigned* __restrict__ keys,
                                                   float* __restrict__ partials) {
    const int ti  = blockIdx.y;
    const int tj  = blockIdx.x;
    const int bid = ti * gridDim.x + tj;
    if (tj < ti) {
        if (threadIdx.x == 0) partials[bid] = 0.0f;
        return;
    }

    __shared__ float ldsA[2][128 * LSTR];
    __shared__ float ldsB[2][128 * LSTR];
    const unsigned baseA[2] = { (unsigned)(uintptr_t)&ldsA[0][0],
                                (unsigned)(uintptr_t)&ldsA[1][0] };
    const unsigned baseB[2] = { (unsigned)(uintptr_t)&ldsB[0][0],
                                (unsigned)(uintptr_t)&ldsB[1][0] };

    const int tid   = threadIdx.x;
    const int lane  = tid & 31;
    const int wave  = tid >> 5;
    const int waveM = wave >> 1;                 // 0..3 -> 32-row strip
    const int waveN = wave & 1;                  // 0..1 -> 64-col strip
    const int rowBase = ti * TILE + waveM * 32;
    const int colBase = tj * TILE + waveN * 64;

    const int r    = lane & 15;
    const int koff = (lane >> 4) * 2;

    // LDS fragment offsets (dwords)
    const int aO0 = (waveM * 32 + r) * LSTR + koff;
    const int aO1 = aO0 + 16 * LSTR;
    const int bO  = (waveN * 64 + r) * LSTR + koff;

    const float* gA = F + (size_t)(ti * TILE) * DK;   // + k0 per panel
    const float* gB = F + (size_t)(tj * TILE) * DK;

    v8f acc[2][4];
#pragma unroll
    for (int mb = 0; mb < 2; ++mb)
#pragma unroll
        for (int nb = 0; nb < 4; ++nb)
#pragma unroll
            for (int v = 0; v < 8; ++v) acc[mb][nb][v] = 0.0f;

    // Prologue: stage panel 0 into buffer 0.
    stage_panel_async(gA, baseA[0], tid);
    stage_panel_async(gB, baseB[0], tid);

    const int NSTEP = DK / KB;                   // 16 panels
    for (int step = 0; step < NSTEP; ++step) {
        const int cur = step & 1;

        if (step + 1 < NSTEP) {                  // kick next panel's DMA
            const int k1 = (step + 1) * KB;
            stage_panel_async(gA + k1, baseA[cur ^ 1], tid);
            stage_panel_async(gB + k1, baseB[cur ^ 1], tid);
            // 4 newly issued per wave; waiting to 8 does not force the previous 4 to land.
            asm volatile("s_wait_asynccnt 0x8" ::: "memory");
        } else {
            asm volatile("s_wait_asynccnt 0x0" ::: "memory");
        }
        __syncthreads();                         // all waves' panels visible

        const float* Ap = &ldsA[cur][0];
        const float* Bp = &ldsB[cur][0];
#pragma unroll
        for (int kk = 0; kk < KB; kk += 4) {
            v2f a0 = *(const v2f*)(Ap + aO0 + kk);
            v2f a1 = *(const v2f*)(Ap + aO1 + kk);
            v2f b0 = *(const v2f*)(Bp + bO + kk);
            v2f b1 = *(const v2f*)(Bp + bO + 16 * LSTR + kk);
            v2f b2 = *(const v2f*)(Bp + bO + 32 * LSTR + kk);
            v2f b3 = *(const v2f*)(Bp + bO + 48 * LSTR + kk);
            // 8 independent accumulators -> no D->A/B RAW hazards.
            acc[0][0] = __builtin_amdgcn_wmma_f32_16x16x4_f32(false, a0, false, b0, (short)0, acc[0][0], false, false);
            acc[0][1] = __builtin_amdgcn_wmma_f32_16x16x4_f32(false, a0, false, b1, (short)0, acc[0][1], false, false);
            acc[0][2] = __builtin_amdgcn_wmma_f32_16x16x4_f32(false, a0, false, b2, (short)0, acc[0][2], false, false);
            acc[0][3] = __builtin_amdgcn_wmma_f32_16x16x4_f32(false, a0, false, b3, (short)0, acc[0][3], false, false);
            acc[1][0] = __builtin_amdgcn_wmma_f32_16x16x4_f32(false, a1, false, b0, (short)0, acc[1][0], false, false);
            acc[1][1] = __builtin_amdgcn_wmma_f32_16x16x4_f32(false, a1, false, b1, (short)0, acc[1][1], false, false);
            acc[1][2] = __builtin_amdgcn_wmma_f32_16x16x4_f32(false, a1, false, b2, (short)0, acc[1][2], false, false);
            acc[1][3] = __builtin_amdgcn_wmma_f32_16x16x4_f32(false, a1, false, b3, (short)0, acc[1][3], false, false);
        }
        __syncthreads();                         // panel consumed; safe to overwrite
    }

    // ---- fused epilogue: gram -> distance -> contrastive term, i<j mask ----
    // C/D layout: n = lane&15, m = v + (lane>>4)*8
    float lsum = 0.0f;
    float    sqj[4];
    unsigned kj[4];
    int      jn[4];
#pragma unroll
    for (int nb = 0; nb < 4; ++nb) {
        int j = colBase + nb * 16 + r;
        jn[nb]  = j;
        sqj[nb] = sq[j];
        kj[nb]  = keys[j];
    }
    const int mSubBase = (lane >> 4) * 8;
#pragma unroll
    for (int mb = 0; mb < 2; ++mb) {
        const int ibase = rowBase + mb * 16 + mSubBase;
#pragma unroll
        for (int v = 0; v < 8; ++v) {
            const int i = ibase + v;
            const float    sqi = sq[i];
            const unsigned ki  = keys[i];
#pragma unroll
            for (int nb = 0; nb < 4; ++nb) {
                float g    = acc[mb][nb][v];
                float d2   = fmaxf(sqi + sqj[nb] - 2.0f * g, 0.0f);
                float dist = sqrtf(d2);
                float t = (ki == kj[nb]) ? (0.5f * dist)
                                         : (0.5f * fmaxf(1.0f - dist, 0.0f));
                if (i < jn[nb]) lsum += t;       // strict upper triangle
            }
        }
    }

    for (int off = 16; off >= 1; off >>= 1) lsum += __shfl_xor(lsum, off, 32);
    __shared__ float red[8];
    if (lane == 0) red[wave] = lsum;
    __syncthreads();
    if (threadIdx.x == 0) {
        float s = 0.0f;
#pragma unroll
        for (int w = 0; w < 8; ++w) s += red[w];
        partials[bid] = s;
    }
}

// ---------------------------------------------------------------------------
// Kernel 3: deterministic fixed-order reduction of 1024 partials.
// ---------------------------------------------------------------------------
__global__ __launch_bounds__(256) void reduce_kernel(const float* __restrict__ partials,
                                                     float* __restrict__ out) {
    __shared__ float red[256];
    float s = 0.0f;
    for (int i = threadIdx.x; i < 1024; i += 256) s += partials[i];
    red[threadIdx.x] = s;
    __syncthreads();
    for (int stride = 128; stride >= 1; stride >>= 1) {
        if ((int)threadIdx.x < stride) red[threadIdx.x] += red[threadIdx.x + stride];
        __syncthreads();
    }
    if (threadIdx.x == 0) out[0] = red[0] * (1.0f / (float)BN);
}

// ---------------------------------------------------------------------------
extern "C" void kernel_launch(void* const* d_in, const int* in_sizes, int n_in,
                              void* d_out, int out_size, void* d_ws, size_t ws_size,
                              hipStream_t stream) {
    (void)in_sizes; (void)n_in; (void)out_size; (void)ws_size;
    const float* F      = (const float*)d_in[0];
    const int*   labels = (const int*)d_in[1];
    float*       out    = (float*)d_out;

    char*     ws       = (char*)d_ws;
    float*    sq       = (float*)ws;               // 4096 floats  (16 KB)
    unsigned* keys     = (unsigned*)(ws + 16384);  // 4096 u32     (16 KB)
    float*    partials = (float*)(ws + 32768);     // 1024 floats  ( 4 KB)

    prep_kernel<<<BN / 8, 256, 0, stream>>>(F, labels, sq, keys);
    dim3 grid(BN / TILE, BN / TILE);               // 32 x 32 tiles
    tile_kernel<<<grid, 256, 0, stream>>>(F, sq, keys, partials);
    reduce_kernel<<<1, 256, 0, stream>>>(partials, out);
}